// model_NN_LSTM_88768384074425
// MI455X (gfx1250) — compile-verified
//
#include <hip/hip_runtime.h>

typedef __attribute__((ext_vector_type(2))) float v2f;
typedef __attribute__((ext_vector_type(8))) float v8f;

#define TILES 2          // independent 16-batch WMMA chains per wave (ILP)
#define HROW  20         // LDS row stride in floats -> conflict-free b64 reloads

// gfx1250 has V_TANH_F32 (TRANS32 op). Prefer the builtin (compiler manages the
// TRANS hazard); fall back to inline asm with an explicit spacer v_nop.
static __device__ __forceinline__ float fast_tanh(float v) {
#if __has_builtin(__builtin_amdgcn_tanhf)
    return __builtin_amdgcn_tanhf(v);
#else
    float r;
    asm volatile("v_tanh_f32 %0, %1\n\tv_nop" : "=v"(r) : "v"(v));
    return r;
#endif
}

__global__ __launch_bounds__(32) void rnn_wmma_f32_kernel(
    const float* __restrict__ x,      // [B, T]
    const float* __restrict__ w_ih,   // [H, 1]
    const float* __restrict__ w_hh,   // [H, H]
    const float* __restrict__ b_ih,   // [H]
    const float* __restrict__ b_hh,   // [H]
    const float* __restrict__ w_lin,  // [1, H]
    const float* __restrict__ b_lin,  // [1]
    float* __restrict__ out,          // [B, T]
    int B, int T)
{
    const int H    = 15;
    const int lane = threadIdx.x;     // 0..31 (wave32)
    const int n    = lane & 15;       // column (hidden) for B/C/D, row (batch) for A
    const int half = lane >> 4;       // 0 / 1
    const int m    = n;               // A-matrix row index held by this lane
    const int batch0 = blockIdx.x * (16 * TILES);

    // ---- B fragments: W_hh^T with w_ih folded into K=15; constant over time.
    // Chunk c covers K = 4c..4c+3; .x holds K=4c+2*half, .y holds K=4c+2*half+1.
    v2f   bv[4];
    float wlx[4], wly[4];
#pragma unroll
    for (int c = 0; c < 4; ++c) {
        const int k0 = 4 * c + 2 * half;
        const int k1 = k0 + 1;
        float e0 = 0.f, e1 = 0.f;
        if (n < H) {
            e0 = (k0 < H) ? w_hh[n * H + k0] : w_ih[n];   // K=15 -> input weight
            e1 = (k1 < H) ? w_hh[n * H + k1] : w_ih[n];
        }
        bv[c].x = e0;
        bv[c].y = e1;
        wlx[c] = (k0 < H) ? w_lin[k0] : 0.f;              // K=15 slot excluded from head
        wly[c] = (k1 < H) ? w_lin[k1] : 0.f;
    }

    // ---- Bias as initial C fragment of chain A; chain B starts from zero.
    const float biasn = (n < H) ? (b_ih[n] + b_hh[n]) : 0.f;
    v8f cbias, czero;
#pragma unroll
    for (int r = 0; r < 8; ++r) { cbias[r] = biasn; czero[r] = 0.f; }
    const float blin = b_lin[0];

    __shared__ float hbuf[TILES][16 * HROW];

    // Per-lane flat offset into x/out: (batch0 + 16*tl + m) * T
    int xoff[TILES];
#pragma unroll
    for (int tl = 0; tl < TILES; ++tl) xoff[tl] = (batch0 + 16 * tl + m) * T;

    // ---- A fragments: h_{-1} = 0, x[·,0] in the K=15 slot (chunk3.y, hi half).
    v2f av[TILES][4];
#pragma unroll
    for (int tl = 0; tl < TILES; ++tl)
#pragma unroll
        for (int c = 0; c < 4; ++c) { av[tl][c].x = 0.f; av[tl][c].y = 0.f; }
#pragma unroll
    for (int tl = 0; tl < TILES; ++tl) {
        const float x0 = x[xoff[tl]];            // all lanes load; hi half keeps it
        if (half) av[tl][3].y = x0;
    }

    for (int t = 0; t < T; ++t) {
        // Prefetch next timestep's input early; consumed only at loop tail.
        // Clamped index: the t==T-1 value lands in an A fragment that is never used.
        const int t1 = (t + 1 < T) ? (t + 1) : (T - 1);
        float xt1[TILES];
#pragma unroll
        for (int tl = 0; tl < TILES; ++tl) xt1[tl] = x[xoff[tl] + t1];

        // D = A(h,x) * B(W_hh^T | w_ih) + bias.  K=16 as two independent
        // 2-deep WMMA chains, merged with packed f32 adds (shorter critical path).
        v8f d[TILES];
#pragma unroll
        for (int tl = 0; tl < TILES; ++tl) {
            v8f accA = cbias;
            v8f accB = czero;
            accA = __builtin_amdgcn_wmma_f32_16x16x4_f32(false, av[tl][0], false, bv[0],
                                                         (short)0, accA, false, false);
            accB = __builtin_amdgcn_wmma_f32_16x16x4_f32(false, av[tl][2], false, bv[2],
                                                         (short)0, accB, false, false);
            accA = __builtin_amdgcn_wmma_f32_16x16x4_f32(false, av[tl][1], false, bv[1],
                                                         (short)0, accA, false, false);
            accB = __builtin_amdgcn_wmma_f32_16x16x4_f32(false, av[tl][3], false, bv[3],
                                                         (short)0, accB, false, false);
            d[tl] = accA + accB;
        }

        // h = tanh(D) via hardware V_TANH_F32; spill to LDS [16][HROW]
        // (row = batch, col = hidden; conflict-free b32 stores).
#pragma unroll
        for (int tl = 0; tl < TILES; ++tl) {
#pragma unroll
            for (int r = 0; r < 8; ++r) {
                const float hv = fast_tanh(d[tl][r]);
                hbuf[tl][(r + 8 * half) * HROW + n] = hv;
            }
        }
        __builtin_amdgcn_wave_barrier();  // LDS ops are in-order per wave; pin scheduling

        // Rebuild A fragments (transpose via LDS, conflict-free b64 loads)
        // + fused output head.
#pragma unroll
        for (int tl = 0; tl < TILES; ++tl) {
            float dot = 0.f;
#pragma unroll
            for (int c = 0; c < 4; ++c) {
                const v2f a = *(const v2f*)&hbuf[tl][m * HROW + 4 * c + 2 * half];
                av[tl][c] = a;
                dot = fmaf(a.x, wlx[c], dot);
                dot = fmaf(a.y, wly[c], dot);
            }
            // next timestep's input into the K=15 slot (single cndmask, no exec edit)
            av[tl][3].y = half ? xt1[tl] : av[tl][3].y;
            // out[b,t] = sum_h h[b,t,h]*w_lin[h] + b_lin : halves hold 8 terms each
            dot += __shfl_xor(dot, 16, 32);
            if (!half)
                out[xoff[tl] + t] = dot + blin;
        }
    }
}

extern "C" void kernel_launch(void* const* d_in, const int* in_sizes, int n_in,
                              void* d_out, int out_size, void* d_ws, size_t ws_size,
                              hipStream_t stream) {
    const float* x     = (const float*)d_in[0];
    const float* w_ih  = (const float*)d_in[1];
    const float* w_hh  = (const float*)d_in[2];
    const float* b_ih  = (const float*)d_in[3];
    const float* b_hh  = (const float*)d_in[4];
    const float* w_lin = (const float*)d_in[5];
    const float* b_lin = (const float*)d_in[6];
    float* out = (float*)d_out;

    const int T = 1024;
    const int B = in_sizes[0] / T;    // x is [B, T, 1]

    dim3 grid(B / (16 * TILES));      // 128 single-wave workgroups
    dim3 block(32);
    hipLaunchKernelGGL(rnn_wmma_f32_kernel, grid, block, 0, stream,
                       x, w_ih, w_hh, b_ih, b_hh, w_lin, b_lin, out, B, T);
}